// PrefetchDenseInstanceNorm_7181185319492
// MI455X (gfx1250) — compile-verified
//
#include <hip/hip_runtime.h>
#include <hip/hip_bf16.h>

// PrefetchDenseInstanceNorm for MI455X (gfx1250).
// Bandwidth-bound (~1.9 GB total traffic -> ~81us floor at 23.3 TB/s).
// Three kernels: (1) per-plane mean/var reduction using V_WMMA_F32_16X16X4_F32
// as the 32-lane wave-reduction primitive, (2) tiny table/window kernel,
// (3) streaming normalize with per-pixel bilinear for the real frame.

typedef __attribute__((ext_vector_type(2))) float v2f;
typedef __attribute__((ext_vector_type(8))) float v8f;

#define N_   5
#define C_   128
#define H_   512
#define W_   512
#define YA_  32
#define XA_  32
#define NPRE (N_ - 1)
#define PLANE_ ((size_t)H_ * (size_t)W_)   // 262144

// Full-wave (32 lane) sum of 64 fp32 partials (2 per lane) via one WMMA:
// D = A(16x4) x ones(4x16); D[m][*] = rowsum(m). Sum of the 8 D VGPRs per lane
// gives half the rows; shfl_xor(16) adds the other half -> every lane holds
// the total. EXEC must be all-1s (guaranteed: called from uniform control flow
// in a fully-populated block).
__device__ __forceinline__ float wave_sum_wmma(float a0, float a1) {
    v2f A; A.x = a0;   A.y = a1;
    v2f B; B.x = 1.0f; B.y = 1.0f;
    v8f Cz = {};
    v8f D = __builtin_amdgcn_wmma_f32_16x16x4_f32(
        /*neg_a=*/false, A, /*neg_b=*/false, B,
        /*c_mod=*/(short)0, Cz, /*reuse_a=*/false, /*reuse_b=*/false);
    float s = ((D[0] + D[1]) + (D[2] + D[3])) + ((D[4] + D[5]) + (D[6] + D[7]));
    s += __shfl_xor(s, 16, 32);
    return s;
}

// ---------------------------------------------------------------------------
// Kernel 1: one block per pre-frame plane (n in 1..4, c in 0..127).
// Computes mean and std (ddof=1) over 512x512 fp32 with float4 streaming.
// ---------------------------------------------------------------------------
__global__ __launch_bounds__(1024)
void reduce_kernel(const float* __restrict__ x,
                   float* __restrict__ pre_mean,  // [NPRE*C_]
                   float* __restrict__ pre_std) { // [NPRE*C_]
    const int p = blockIdx.x;                       // 0..NPRE*C_-1
    const float4* __restrict__ src =
        (const float4*)(x + ((size_t)(C_ + p)) * PLANE_);

    float sx = 0.f, sy = 0.f, qx = 0.f, qy = 0.f;
    const int nvec = (int)(PLANE_ / 4);             // 65536 float4s
    for (int i = threadIdx.x; i < nvec; i += 1024) {
        float4 v = src[i];
        sx += v.x + v.y;
        sy += v.z + v.w;
        qx = fmaf(v.x, v.x, qx); qx = fmaf(v.y, v.y, qx);
        qy = fmaf(v.z, v.z, qy); qy = fmaf(v.w, v.w, qy);
    }

    // Wave-level reduction via WMMA (uniform control flow, EXEC all-1s).
    float wsum = wave_sum_wmma(sx, sy);
    float wsq  = wave_sum_wmma(qx, qy);

    __shared__ float red[64];
    const int wave = threadIdx.x >> 5;
    if ((threadIdx.x & 31) == 0) { red[wave] = wsum; red[32 + wave] = wsq; }
    __syncthreads();

    if (threadIdx.x == 0) {
        float S = 0.f, Q = 0.f;
        for (int i = 0; i < 32; ++i) { S += red[i]; Q += red[32 + i]; }
        const float M   = (float)PLANE_;
        const float mu  = S / M;
        float var = (Q - S * S / M) / (M - 1.0f);   // ddof = 1
        var = var > 0.f ? var : 0.f;
        pre_mean[p] = mu;
        pre_std[p]  = sqrtf(var);
    }
}

// ---------------------------------------------------------------------------
// Kernel 2: per-channel 3x3 window build. 1 block x 128 threads.
// Applies anchor scatter (last-write-wins over i), edge clamp (== edge pad),
// zero -> center replacement.
// ---------------------------------------------------------------------------
__global__ __launch_bounds__(C_)
void tables_kernel(const float* __restrict__ mean_table, // [YA,XA,C]
                   const float* __restrict__ std_table,
                   const int* __restrict__ pya, const int* __restrict__ pxa,
                   const int* __restrict__ yanch, const int* __restrict__ xanch,
                   const float* __restrict__ pre_mean,
                   const float* __restrict__ pre_std,
                   float* __restrict__ m33ws,   // [C*9]
                   float* __restrict__ s33ws,   // [C*9]
                   int npre) {
    const int c  = threadIdx.x;
    const int ya = yanch[0];
    const int xa = xanch[0];

    float m[9], s[9];
    #pragma unroll
    for (int p = 0; p < 9; ++p) {
        const int dy = p / 3, dx = p % 3;
        int yy = ya - 1 + dy; yy = yy < 0 ? 0 : (yy > YA_ - 1 ? YA_ - 1 : yy);
        int xx = xa - 1 + dx; xx = xx < 0 ? 0 : (xx > XA_ - 1 ? XA_ - 1 : xx);
        float mv = mean_table[((size_t)yy * XA_ + xx) * C_ + c];
        float sv = std_table [((size_t)yy * XA_ + xx) * C_ + c];
        for (int i = 0; i < npre; ++i) {
            const int py = pya[i];
            if (py >= 0 && py == yy && pxa[i] == xx) {   // valid scatter hit
                mv = pre_mean[i * C_ + c];
                sv = pre_std [i * C_ + c];
            }
        }
        m[p] = mv; s[p] = sv;
    }
    const float cm = m[4], cs = s[4];
    #pragma unroll
    for (int p = 0; p < 9; ++p) {
        m33ws[c * 9 + p] = (m[p] == 0.0f) ? cm : m[p];
        s33ws[c * 9 + p] = (s[p] == 0.0f) ? cs : s[p];
    }
}

// ---------------------------------------------------------------------------
// Kernel 3: streaming normalize. Each block = 2 rows (same n,c), 256 threads,
// one float4 per thread in and out.
// ---------------------------------------------------------------------------
__global__ __launch_bounds__(256)
void apply_kernel(const float* __restrict__ x,
                  const float* __restrict__ weight, const float* __restrict__ bias,
                  const float* __restrict__ pre_mean, const float* __restrict__ pre_std,
                  const float* __restrict__ m33ws, const float* __restrict__ s33ws,
                  float* __restrict__ out) {
    const int r0   = blockIdx.x * 2;            // first row of this block
    const int n    = r0 / (C_ * H_);            // uniform across block
    const int rem  = r0 % (C_ * H_);
    const int c    = rem / H_;
    const int h0   = rem % H_;
    const int lrow = threadIdx.x >> 7;          // 0..1
    const int col4 = threadIdx.x & 127;         // 0..127 (float4 index in row)
    const int h    = h0 + lrow;

    const size_t base = (size_t)(r0 + lrow) * W_;
    const float4* __restrict__ src = (const float4*)(x + base);
    float4* __restrict__ dst       = (float4*)(out + base);

    const float wgt = weight[c];
    const float bs  = bias[c];

    __shared__ float sm[9], ss[9];

    float4 v = src[col4];
    float4 o;

    if (n == 0) {   // uniform branch: __syncthreads is safe
        if (threadIdx.x < 9)       sm[threadIdx.x]     = m33ws[c * 9 + threadIdx.x];
        else if (threadIdx.x < 18) ss[threadIdx.x - 9] = s33ws[c * 9 + (threadIdx.x - 9)];
        __syncthreads();

        const float ty = 0.5f + ((float)h + 0.5f) * (1.0f / (float)H_);
        int y0 = (int)floorf(ty); y0 = y0 < 0 ? 0 : (y0 > 1 ? 1 : y0);
        const float wy  = ty - (float)y0;
        const float omy = 1.0f - wy;

        #pragma unroll
        for (int j = 0; j < 4; ++j) {
            const int w = col4 * 4 + j;
            const float tx = 0.5f + ((float)w + 0.5f) * (1.0f / (float)W_);
            int x0 = (int)floorf(tx); x0 = x0 < 0 ? 0 : (x0 > 1 ? 1 : x0);
            const float wx  = tx - (float)x0;
            const float omx = 1.0f - wx;
            const int i00 = y0 * 3 + x0;

            const float mtop = sm[i00]     * omx + sm[i00 + 1] * wx;
            const float mbot = sm[i00 + 3] * omx + sm[i00 + 4] * wx;
            const float mv   = mtop * omy + mbot * wy;

            const float stop = ss[i00]     * omx + ss[i00 + 1] * wx;
            const float sbot = ss[i00 + 3] * omx + ss[i00 + 4] * wx;
            const float sv   = stop * omy + sbot * wy;
            const float inv  = 1.0f / sv;   // reciprocal AFTER interpolation (matches ref)

            const float xi = (&v.x)[j];
            (&o.x)[j] = (xi - mv) * inv * wgt + bs;
        }
    } else {
        const float mu = pre_mean[(n - 1) * C_ + c];
        const float sd = pre_std [(n - 1) * C_ + c];
        #pragma unroll
        for (int j = 0; j < 4; ++j) {
            const float xi = (&v.x)[j];
            (&o.x)[j] = (xi - mu) / sd * wgt + bs;   // division, matching reference
        }
    }
    dst[col4] = o;
}

// ---------------------------------------------------------------------------
extern "C" void kernel_launch(void* const* d_in, const int* in_sizes, int n_in,
                              void* d_out, int out_size, void* d_ws, size_t ws_size,
                              hipStream_t stream) {
    const float* x          = (const float*)d_in[0];
    const float* mean_table = (const float*)d_in[1];
    const float* std_table  = (const float*)d_in[2];
    const float* weight     = (const float*)d_in[3];
    const float* bias       = (const float*)d_in[4];
    const int*   pya        = (const int*)d_in[5];
    const int*   pxa        = (const int*)d_in[6];
    const int*   yanch      = (const int*)d_in[7];
    const int*   xanch      = (const int*)d_in[8];
    float*       out        = (float*)d_out;

    const int npre = in_sizes[5];   // = NPRE = 4

    // Workspace layout (floats): pre_mean[512] | pre_std[512] | m33[1152] | s33[1152]
    float* ws       = (float*)d_ws;
    float* pre_mean = ws;
    float* pre_std  = ws + NPRE * C_;
    float* m33ws    = ws + 2 * NPRE * C_;
    float* s33ws    = m33ws + C_ * 9;

    // 1) per-plane mean/std over pre frames
    reduce_kernel<<<NPRE * C_, 1024, 0, stream>>>(x, pre_mean, pre_std);

    // 2) table scatter + 3x3 window
    tables_kernel<<<1, C_, 0, stream>>>(mean_table, std_table, pya, pxa,
                                        yanch, xanch, pre_mean, pre_std,
                                        m33ws, s33ws, npre);

    // 3) streaming normalize: 2 rows per block
    const int nblocks = (N_ * C_ * H_) / 2;     // 163840
    apply_kernel<<<nblocks, 256, 0, stream>>>(x, weight, bias,
                                              pre_mean, pre_std,
                                              m33ws, s33ws, out);
}